// MyModel_87522843560215
// MI455X (gfx1250) — compile-verified
//
#include <hip/hip_runtime.h>
#include <hip/hip_bf16.h>
#include <math.h>

// ---------------- problem constants ----------------
#define B_   16
#define T_   20
#define H_   96
#define W_   96
#define IC_  3
#define F_   4          // FILTERS
#define G_   16         // 4*FILTERS gate channels
#define HP_  95
#define WP_  95
#define HS_  96         // padded h-state stride (pad row/col are zeros)
#define NPIX (HP_ * WP_)          // 9025
#define PH_  47
#define PW_  47
#define KIN  12                   // 2*2*3
#define KREC 16                   // 2*2*4
#define KTOT 28                   // fused K (input-conv + recurrent-conv)
#define WAVES_PER_BLOCK 8
#define TILES      ((NPIX + 15) / 16)                                 // 565
#define TILE_BLOCKS ((TILES + WAVES_PER_BLOCK - 1) / WAVES_PER_BLOCK) // 71

#define SH  ((size_t)B_ * HS_ * HS_ * F_)   // padded h buffer floats (589,824)
#define SC  ((size_t)B_ * NPIX * F_)        // cell state floats      (577,600)

typedef __attribute__((ext_vector_type(2))) float v2f;
typedef __attribute__((ext_vector_type(8))) float v8f;

// A-matrix element: fused K row for one output pixel (y,x). Branch on k is
// wave-uniform per unrolled step (k-blocks of 4 never straddle the 12 split).
// Recurrent reads are unconditionally in-bounds thanks to the padded h layout
// (pad row/col hold the 'SAME'-padding zeros).
__device__ __forceinline__ float a_elem(const float* __restrict__ in_bt,
                                        const float* __restrict__ hp_b,
                                        int y, int x, int k)
{
    if (k < KIN) {
        int kh = k / 6;
        int kw = (k / 3) & 1;
        int ic = k % 3;
        return in_bt[((size_t)(y + kh) * W_ + (x + kw)) * IC_ + ic];
    } else {
        int kk = k - KIN;
        int kh = kk >> 3;
        int kw = (kk >> 2) & 1;
        int fc = kk & 3;
        return hp_b[((size_t)(y + kh) * HS_ + (x + kw)) * F_ + fc];
    }
}

// B-matrix element: stacked weights, row k, gate-channel n.
__device__ __forceinline__ float w_elem(const float* __restrict__ kern,
                                        const float* __restrict__ rkern,
                                        int k, int n)
{
    return (k < KIN) ? kern[k * G_ + n] : rkern[(k - KIN) * G_ + n];
}

__device__ __forceinline__ float sigmoidf(float v) { return 1.0f / (1.0f + expf(-v)); }

// One ConvLSTM timestep: z = [x*Wk + h_prev*Wr + b] via fused implicit GEMM
// (M=16 pixels, N=16 gates, K=28) using V_WMMA_F32_16X16X4_F32, then gates.
__global__ __launch_bounds__(256) void convlstm_step(
    const float* __restrict__ inp,     // (B,T,H,W,IC)
    const float* __restrict__ kern,    // (2,2,3,16) flat
    const float* __restrict__ rkern,   // (2,2,4,16) flat
    const float* __restrict__ bias,    // (16)
    const float* __restrict__ h_prev,  // (B,HS,HS,F) padded, pad = 0
    float* __restrict__ h_cur,         // (B,HS,HS,F) padded
    float* __restrict__ c_state,       // (B,NPIX,F)
    int t)
{
    __shared__ float zt[WAVES_PER_BLOCK][16][16];

    const int lane = threadIdx.x & 31;
    const int wave = threadIdx.x >> 5;
    const int b    = blockIdx.x / TILE_BLOCKS;
    const int tile_raw = (blockIdx.x % TILE_BLOCKS) * WAVES_PER_BLOCK + wave;
    const int tile = (tile_raw < TILES) ? tile_raw : (TILES - 1);

    const bool hi = lane >= 16;
    const int  n  = lane & 15;                 // gate channel (B/C column)
    const int  m  = lane & 15;                 // pixel row within tile (A row)

    const int p_raw = tile * 16 + m;
    const int p     = (p_raw < NPIX) ? p_raw : (NPIX - 1);   // clamp for loads only
    const int y     = p / WP_;
    const int x     = p % WP_;

    const float* in_bt = inp    + ((size_t)b * T_ + t) * (size_t)(H_ * W_ * IC_);
    const float* hp_b  = h_prev + (size_t)b * HS_ * HS_ * F_;

    // Pull next timestep's input slice toward L2 while this step computes
    // (t is uniform -> scalar branch; emits global_prefetch_b8).
    if (t + 1 < T_) {
        __builtin_prefetch(in_bt + (size_t)(H_ * W_ * IC_) +
                               ((size_t)y * W_ + x) * IC_, 0, 1);
    }

    // Accumulator pre-loaded with bias (z = A*B + bias through C operand).
    v8f acc;
    {
        float bn = bias[n];
#pragma unroll
        for (int r = 0; r < 8; ++r) acc[r] = bn;
    }

    // 7 chained K=4 f32 WMMAs, fully branch-free (EXEC all-1s throughout).
    //  A 16x4: lanes0-15 VGPR{0,1}=K{0,1}; lanes16-31 VGPR{0,1}=K{2,3}
    //  B  4x16: VGPR r -> K = r (lanes0-15) / K = 2+r (lanes16-31), N = lane%16
#pragma unroll
    for (int s = 0; s < KTOT / 4; ++s) {
        const int kb = 4 * s + (hi ? 2 : 0);
        v2f a, bm;
        a.x  = a_elem(in_bt, hp_b, y, x, kb);
        a.y  = a_elem(in_bt, hp_b, y, x, kb + 1);
        bm.x = w_elem(kern, rkern, kb,     n);
        bm.y = w_elem(kern, rkern, kb + 1, n);
        acc = __builtin_amdgcn_wmma_f32_16x16x4_f32(
                  false, a, false, bm, (short)0, acc, false, false);
    }

    // Spill z tile (C/D layout: VGPR r -> row r (lanes0-15) / row r+8 (lanes16-31)).
    {
        const int mb = hi ? 8 : 0;
#pragma unroll
        for (int r = 0; r < 8; ++r) zt[wave][mb + r][n] = acc[r];
    }
    __syncthreads();

    float* hc_b = h_cur   + (size_t)b * HS_ * HS_ * F_;
    float* cs_b = c_state + (size_t)b * NPIX * F_;

    // 16 pixels x 4 filters = 64 gate updates per wave -> 2 per lane.
#pragma unroll
    for (int j = 0; j < 2; ++j) {
        const int idx = lane * 2 + j;
        const int mm  = idx >> 2;
        const int f   = idx & 3;
        const int pp  = tile_raw * 16 + mm;
        if (tile_raw < TILES && pp < NPIX) {
            float zi = zt[wave][mm][f];
            float zf = zt[wave][mm][4 + f];
            float zg = zt[wave][mm][8 + f];
            float zo = zt[wave][mm][12 + f];
            float ig = sigmoidf(zi);
            float fg = sigmoidf(zf);
            float gg = tanhf(zg);
            float og = sigmoidf(zo);
            const int py = pp / WP_;
            const int px = pp % WP_;
            size_t offc = (size_t)pp * F_ + f;
            float cn = fg * cs_b[offc] + ig * gg;
            cs_b[offc] = cn;
            hc_b[((size_t)py * HS_ + px) * F_ + f] = og * tanhf(cn);
        }
    }
}

// MaxPool(1,2,2) + dense-dot contribution of timestep t, folded into out[b].
__global__ __launch_bounds__(256) void pool_dense(
    const float* __restrict__ h,       // (B,HS,HS,F) padded, this timestep
    const float* __restrict__ dw,      // (T*PH*PW*F, 1)
    float* __restrict__ out,           // (B,)
    int t)
{
    __shared__ float red[256];
    const int b = blockIdx.x;
    const float* hb = h + (size_t)b * HS_ * HS_ * F_;

    float sum = 0.0f;
    for (int i = threadIdx.x; i < PH_ * PW_ * F_; i += blockDim.x) {
        int f  = i & 3;
        int pw = (i >> 2) % PW_;
        int ph = i / (4 * PW_);
        int y = 2 * ph, x = 2 * pw;
        float v =    hb[((size_t)y * HS_ + x) * F_ + f];
        v = fmaxf(v, hb[((size_t)y * HS_ + x + 1) * F_ + f]);
        v = fmaxf(v, hb[((size_t)(y + 1) * HS_ + x) * F_ + f]);
        v = fmaxf(v, hb[((size_t)(y + 1) * HS_ + x + 1) * F_ + f]);
        int flat = ((t * PH_ + ph) * PW_ + pw) * F_ + f;
        sum += v * dw[flat];
    }
    red[threadIdx.x] = sum;
    __syncthreads();
    for (int s = 128; s > 0; s >>= 1) {
        if (threadIdx.x < s) red[threadIdx.x] += red[threadIdx.x + s];
        __syncthreads();
    }
    if (threadIdx.x == 0) out[b] += red[0];   // one block per b per t; stream-ordered
}

// Zero the whole state region (both padded h buffers + cell state) and seed out
// with the dense bias. Zeroing both h buffers also establishes the permanent
// zero pad row/col that implements the recurrent conv's 'SAME' padding.
__global__ void init_state(float* __restrict__ ws, float* __restrict__ out,
                           const float* __restrict__ db, size_t nstate)
{
    size_t i = (size_t)blockIdx.x * blockDim.x + threadIdx.x;
    if (i < nstate) ws[i] = 0.0f;
    if (i < B_) out[i] = db[0];
}

extern "C" void kernel_launch(void* const* d_in, const int* in_sizes, int n_in,
                              void* d_out, int out_size, void* d_ws, size_t ws_size,
                              hipStream_t stream)
{
    const float* inp   = (const float*)d_in[0];  // (B,T,H,W,IC)
    const float* kern  = (const float*)d_in[1];  // (2,2,3,16)
    const float* rkern = (const float*)d_in[2];  // (2,2,4,16)
    const float* bias  = (const float*)d_in[3];  // (16)
    const float* dw    = (const float*)d_in[4];  // (FLAT,1)
    const float* db    = (const float*)d_in[5];  // (1)
    float* out = (float*)d_out;                  // (B,1) fp32

    float* hA = (float*)d_ws;                    // padded h double-buffer A
    float* hB = hA + SH;                         // padded h double-buffer B
    float* cS = hB + SH;                         // cell state (compact)

    const size_t nstate = 2 * SH + SC;           // 1,757,248 floats (~7 MB)
    init_state<<<dim3((unsigned)((nstate + 255) / 256)), dim3(256), 0, stream>>>(
        (float*)d_ws, out, db, nstate);

    float* hp = hA;
    float* hc = hB;
    for (int t = 0; t < T_; ++t) {
        convlstm_step<<<dim3(B_ * TILE_BLOCKS), dim3(256), 0, stream>>>(
            inp, kern, rkern, bias, hp, hc, cS, t);
        pool_dense<<<dim3(B_), dim3(256), 0, stream>>>(hc, dw, out, t);
        float* tmp = hp; hp = hc; hc = tmp;
    }
}